// SageLayer_71038759076173
// MI455X (gfx1250) — compile-verified
//
#include <hip/hip_runtime.h>
#include <hip/hip_bf16.h>
#include <math.h>

// ---------------------------------------------------------------------------
// GraphSAGE layer for MI455X (gfx1250, wave32).
//
// Algebraic folding: mean commutes with the affine neighbor transform, and the
// 128->128 projections + 384->384 WC linear collapse into a single 384x384
// matrix M (precomputed on device) and bias vector c. The remaining heavy ops:
//   (1) neighbor-mean gathers (bandwidth bound, ~0.54 GB HBM random rows)
//   (2) one [16384,384]x[384,384] fp32 GEMM  -> V_WMMA_F32_16X16X4_F32
//   (3) fused leaky-relu + L2-normalize epilogue
//   (4) final [32768] row gather
// ---------------------------------------------------------------------------

typedef __attribute__((ext_vector_type(2))) float v2f;
typedef __attribute__((ext_vector_type(8))) float v8f;

#define U_NODES 16384
#define KNBR    32
#define DIN     128
#define DOUT    384
#define BATCH   32768
#define XPAD    388   // 388 % 64 == 4 -> conflict-free LDS column walks

// ------------------------- weight folding, stage 1 -------------------------
// Cadj = W_adj @ Wa_adj, Cdis = W_dis @ Wa_dis, bb* = W_* @ ba_*
__global__ void precompute1(const float* __restrict__ Wa_adj, const float* __restrict__ ba_adj,
                            const float* __restrict__ Wa_dis, const float* __restrict__ ba_dis,
                            const float* __restrict__ W_adj,  const float* __restrict__ W_dis,
                            float* __restrict__ Cadj, float* __restrict__ Cdis,
                            float* __restrict__ bbadj, float* __restrict__ bbdis) {
    int tid = blockIdx.x * blockDim.x + threadIdx.x;
    if (tid < 16384) {
        int j = tid >> 7, d = tid & 127;
        float s = 0.f;
        for (int t = 0; t < 128; ++t) s = fmaf(W_adj[j * 128 + t], Wa_adj[t * 128 + d], s);
        Cadj[tid] = s;
    } else if (tid < 32768) {
        int r = tid - 16384, j = r >> 7, d = r & 127;
        float s = 0.f;
        for (int t = 0; t < 128; ++t) s = fmaf(W_dis[j * 128 + t], Wa_dis[t * 128 + d], s);
        Cdis[r] = s;
    } else if (tid < 32896) {
        int j = tid - 32768;
        float s = 0.f;
        for (int t = 0; t < 128; ++t) s = fmaf(W_adj[j * 128 + t], ba_adj[t], s);
        bbadj[j] = s;
    } else if (tid < 33024) {
        int j = tid - 32896;
        float s = 0.f;
        for (int t = 0; t < 128; ++t) s = fmaf(W_dis[j * 128 + t], ba_dis[t], s);
        bbdis[j] = s;
    }
}

// ------------------------- weight folding, stage 2 -------------------------
// Mt[d][o] (K-major!) so GEMM B-fragment loads are coalesced rows.
//   d<128   : sum_j WC[o][j]       * W_self[j][d]
//   d<256   : sum_j WC[o][128+j]   * Cadj[j][d-128]
//   else    : sum_j WC[o][256+j]   * Cdis[j][d-256]
// c[o] = bWC[o] + bias[o] + WC_mid@bbadj + WC_hi@bbdis
__global__ void precompute2(const float* __restrict__ WC, const float* __restrict__ bWC,
                            const float* __restrict__ bias, const float* __restrict__ W_self,
                            const float* __restrict__ Cadj, const float* __restrict__ Cdis,
                            const float* __restrict__ bbadj, const float* __restrict__ bbdis,
                            float* __restrict__ Mt, float* __restrict__ cvec) {
    int tid = blockIdx.x * blockDim.x + threadIdx.x;
    if (tid < DOUT * DOUT) {
        int d = tid / DOUT, o = tid % DOUT;
        float s = 0.f;
        if (d < 128) {
            for (int j = 0; j < 128; ++j) s = fmaf(WC[o * DOUT + j], W_self[j * 128 + d], s);
        } else if (d < 256) {
            int dd = d - 128;
            for (int j = 0; j < 128; ++j) s = fmaf(WC[o * DOUT + 128 + j], Cadj[j * 128 + dd], s);
        } else {
            int dd = d - 256;
            for (int j = 0; j < 128; ++j) s = fmaf(WC[o * DOUT + 256 + j], Cdis[j * 128 + dd], s);
        }
        Mt[d * DOUT + o] = s;
    } else if (tid < DOUT * DOUT + DOUT) {
        int o = tid - DOUT * DOUT;
        float s = bWC[o] + bias[o];
        for (int j = 0; j < 128; ++j) {
            s = fmaf(WC[o * DOUT + 128 + j], bbadj[j], s);
            s = fmaf(WC[o * DOUT + 256 + j], bbdis[j], s);
        }
        cvec[o] = s;
    }
}

// ------------------------------- aggregation -------------------------------
// One wave32 per node; lane l owns float4 slice [4l,4l+4). Each embedding row
// is one contiguous 512B line -> perfectly coalesced wave-level gather.
// X row layout: [ self(128) | mean_adj(128) | mean_dis(128) ]
__global__ __launch_bounds__(256) void aggregate(const float* __restrict__ emb,
                                                 const int* __restrict__ unique_ids,
                                                 const int* __restrict__ adj_neigh,
                                                 const int* __restrict__ dis_neigh,
                                                 float* __restrict__ X) {
    const int wave = threadIdx.x >> 5;
    const int lane = threadIdx.x & 31;
    const int u = blockIdx.x * 8 + wave;
    const float4* emb4 = (const float4*)emb;

    const int ns = unique_ids[u];
    float4 sv = emb4[(size_t)ns * 32 + lane];

    float4 aa = make_float4(0.f, 0.f, 0.f, 0.f);
    float4 ad = make_float4(0.f, 0.f, 0.f, 0.f);
    for (int k = 0; k < KNBR; ++k) {          // wave-uniform index -> scalar load
        int n = adj_neigh[u * KNBR + k];
        float4 e = emb4[(size_t)n * 32 + lane];
        aa.x += e.x; aa.y += e.y; aa.z += e.z; aa.w += e.w;
    }
    for (int k = 0; k < KNBR; ++k) {
        int n = dis_neigh[u * KNBR + k];
        float4 e = emb4[(size_t)n * 32 + lane];
        ad.x += e.x; ad.y += e.y; ad.z += e.z; ad.w += e.w;
    }
    const float inv = 1.0f / (float)KNBR;
    float4* X4 = (float4*)X;
    size_t base = (size_t)u * (DOUT / 4);
    X4[base + lane] = sv;
    X4[base + 32 + lane] = make_float4(aa.x * inv, aa.y * inv, aa.z * inv, aa.w * inv);
    X4[base + 64 + lane] = make_float4(ad.x * inv, ad.y * inv, ad.z * inv, ad.w * inv);
}

// --------------------------- WMMA GEMM + epilogue --------------------------
// F[16384,384] = l2norm(leakyrelu(X @ Mt + c)). 16 rows per block, 8 waves x
// 48 cols each (3 16x16 N-tiles). K stepped by 4 with V_WMMA_F32_16X16X4_F32.
// Fragment layouts per ISA 7.12.2:
//   A 16x4 f32 : lanes 0-15 -> M=lane, VGPR{0,1}=K{0,1}; lanes 16-31 -> K{2,3}
//   B 4x16 f32 : assumed symmetric striping (lanes 0-15 N=lane K={0,1},
//                lanes 16-31 N=lane-16 K={2,3}) - compile-only, unverifiable.
//   C/D 16x16  : VGPR r -> M=r (lanes 0-15) / M=8+r (lanes 16-31), N=lane&15
__global__ __launch_bounds__(256) void gemm_norm(const float* __restrict__ X,
                                                 const float* __restrict__ Mt,
                                                 const float* __restrict__ cvec,
                                                 float* __restrict__ F) {
    __shared__ float Xs[16 * XPAD];
    __shared__ float ninv[16];

    const int tid = threadIdx.x;
    const int r0 = blockIdx.x * 16;

    for (int i = tid; i < 16 * DOUT; i += 256) {
        int r = i / DOUT, c = i % DOUT;
        Xs[r * XPAD + c] = X[(size_t)(r0 + r) * DOUT + c];
    }
    __syncthreads();

    const int wave = tid >> 5;
    const int lane = tid & 31;
    const int laneN = lane & 15;
    const int half = lane >> 4;
    const int nbase = wave * 48;

    v8f acc0 = {}; v8f acc1 = {}; v8f acc2 = {};

    for (int k0 = 0; k0 < DOUT; k0 += 4) {
        const int ka = k0 + 2 * half;
        v2f a;
        a.x = Xs[laneN * XPAD + ka];
        a.y = Xs[laneN * XPAD + ka + 1];
        const float* b0r = Mt + (size_t)ka * DOUT;
        const float* b1r = b0r + DOUT;

        v2f b;
        b.x = b0r[nbase + laneN];      b.y = b1r[nbase + laneN];
        acc0 = __builtin_amdgcn_wmma_f32_16x16x4_f32(false, a, false, b, (short)0, acc0, false, false);
        b.x = b0r[nbase + 16 + laneN]; b.y = b1r[nbase + 16 + laneN];
        acc1 = __builtin_amdgcn_wmma_f32_16x16x4_f32(false, a, false, b, (short)0, acc1, false, false);
        b.x = b0r[nbase + 32 + laneN]; b.y = b1r[nbase + 32 + laneN];
        acc2 = __builtin_amdgcn_wmma_f32_16x16x4_f32(false, a, false, b, (short)0, acc2, false, false);
    }
    __syncthreads();   // all waves done reading Xs -> reuse as output tile

#pragma unroll
    for (int e = 0; e < 8; ++e) {
        const int row = half * 8 + e;
        int n = nbase + laneN;
        float v = acc0[e] + cvec[n]; v = v > 0.f ? v : 0.2f * v; Xs[row * XPAD + n] = v;
        n += 16;
        v = acc1[e] + cvec[n]; v = v > 0.f ? v : 0.2f * v; Xs[row * XPAD + n] = v;
        n += 16;
        v = acc2[e] + cvec[n]; v = v > 0.f ? v : 0.2f * v; Xs[row * XPAD + n] = v;
    }
    __syncthreads();

    if (tid < 16) {
        float s = 0.f;
        for (int j = 0; j < DOUT; ++j) { float v = Xs[tid * XPAD + j]; s = fmaf(v, v, s); }
        ninv[tid] = 1.0f / fmaxf(sqrtf(s), 1e-12f);
    }
    __syncthreads();

    for (int i = tid; i < 16 * DOUT; i += 256) {
        int r = i / DOUT, c = i % DOUT;
        F[(size_t)(r0 + r) * DOUT + c] = Xs[r * XPAD + c] * ninv[r];
    }
}

// ------------------------------ final gather -------------------------------
__global__ void gather_out(const float* __restrict__ F, const int* __restrict__ nodes_idx,
                           float* __restrict__ out) {
    int i = blockIdx.x * blockDim.x + threadIdx.x;   // one float4 per thread
    if (i >= BATCH * (DOUT / 4)) return;
    int b = i / (DOUT / 4), q = i % (DOUT / 4);
    int u = nodes_idx[b];
    ((float4*)out)[i] = ((const float4*)F)[(size_t)u * (DOUT / 4) + q];
}

// ---------------------------------------------------------------------------
extern "C" void kernel_launch(void* const* d_in, const int* in_sizes, int n_in,
                              void* d_out, int out_size, void* d_ws, size_t ws_size,
                              hipStream_t stream) {
    const float* emb        = (const float*)d_in[0];
    const float* Wa_adj     = (const float*)d_in[1];
    const float* ba_adj     = (const float*)d_in[2];
    const float* Wa_dis     = (const float*)d_in[3];
    const float* ba_dis     = (const float*)d_in[4];
    const float* W_self     = (const float*)d_in[5];
    const float* W_adj      = (const float*)d_in[6];
    const float* W_dis      = (const float*)d_in[7];
    const float* WC         = (const float*)d_in[8];
    const float* bWC        = (const float*)d_in[9];
    const float* bias       = (const float*)d_in[10];
    const int*   unique_ids = (const int*)d_in[11];
    const int*   adj_neigh  = (const int*)d_in[12];
    const int*   dis_neigh  = (const int*)d_in[13];
    const int*   nodes_idx  = (const int*)d_in[14];

    float* ws    = (float*)d_ws;
    float* X     = ws;                              // 16384*384
    float* F     = X + (size_t)U_NODES * DOUT;      // 16384*384
    float* Mt    = F + (size_t)U_NODES * DOUT;      // 384*384
    float* cvec  = Mt + DOUT * DOUT;                // 384
    float* Cadj  = cvec + DOUT;                     // 128*128
    float* Cdis  = Cadj + 16384;                    // 128*128
    float* bbadj = Cdis + 16384;                    // 128
    float* bbdis = bbadj + 128;                     // 128

    precompute1<<<130, 256, 0, stream>>>(Wa_adj, ba_adj, Wa_dis, ba_dis, W_adj, W_dis,
                                         Cadj, Cdis, bbadj, bbdis);
    precompute2<<<578, 256, 0, stream>>>(WC, bWC, bias, W_self, Cadj, Cdis, bbadj, bbdis,
                                         Mt, cvec);
    aggregate<<<U_NODES / 8, 256, 0, stream>>>(emb, unique_ids, adj_neigh, dis_neigh, X);
    gemm_norm<<<U_NODES / 16, 256, 0, stream>>>(X, Mt, cvec, F);
    gather_out<<<(BATCH * (DOUT / 4)) / 256, 256, 0, stream>>>(F, nodes_idx, (float*)d_out);
}